// DynMorphConv2d_28527172780064
// MI455X (gfx1250) — compile-verified
//
#include <hip/hip_runtime.h>
#include <hip/hip_bf16.h>
#include <math.h>

// Problem constants (from reference): x[B,C,H,W], 3x3 max-plus stencil.
#define B_   8
#define C_   192
#define H_   128
#define W_   128
#define TH   32            // output rows per strip
#define TROWS (TH + 2)     // staged rows incl. top/bottom halo

// ---- gfx1250 async global->LDS (ASYNCcnt) ------------------------------
__device__ __forceinline__ void async_load_b128_to_lds(uint32_t lds_addr,
                                                       const void* gaddr) {
  // Per active lane: copies 16 bytes from global address gaddr to LDS address lds_addr (ASYNCcnt++)
  asm volatile("global_load_async_to_lds_b128 %0, %1, off"
               :: "v"(lds_addr), "v"(gaddr)
               : "memory");
}

__device__ __forceinline__ void wait_async0() {
  asm volatile("s_wait_asynccnt 0" ::: "memory");
}

__global__ __launch_bounds__(256)
void dynmorph_kernel(const float* __restrict__ x,
                     const float* __restrict__ kern,
                     const float* __restrict__ gate_w,
                     const float* __restrict__ gate_b,
                     float* __restrict__ out) {
  __shared__ __align__(16) float tile[TROWS * W_];   // 34 x 128 floats = 17408 B

  const int strip = blockIdx.x;            // 0..3
  const int plane = blockIdx.y;            // 0..B*C-1
  const int c     = plane % C_;
  const int h0    = strip * TH;
  const int tid   = (int)threadIdx.x;

  const float* __restrict__ xp = x + (size_t)plane * (size_t)(H_ * W_);

  // ---- Stage 34 rows x 128 floats into LDS via async tensor-data path ----
  // 32 x b128 chunks per row; lanes of one wave cover exactly one row -> no
  // divergence in the boundary test, fully coalesced 512B per wave per row.
  const uint32_t lds_base = (uint32_t)(uintptr_t)(&tile[0]);
  for (int q = tid; q < TROWS * (W_ / 4); q += 256) {
    const int t = q >> 5;                 // tile row (32 chunks per row)
    const int j = (q & 31) << 2;          // first float of this 16B chunk
    const int h = h0 - 1 + t;             // source image row (with halo)
    if (h >= 0 && h < H_) {
      async_load_b128_to_lds(lds_base + (uint32_t)((t * W_ + j) * 4),
                             (const void*)(xp + h * W_ + j));
    } else {
      // zero padding rows (reference uses zero-pad): plain DS store
      *(float4*)(&tile[t * W_ + j]) = make_float4(0.f, 0.f, 0.f, 0.f);
    }
  }
  wait_async0();        // ASYNCcnt -> 0 (barrier only waits on DScnt)
  __syncthreads();

  // ---- Per-channel uniform parameters (scalarizable loads) ----
  float kv[9];
#pragma unroll
  for (int v = 0; v < 9; ++v) kv[v] = kern[c * 9 + v];
  const float gw = gate_w[c];
  const float gb = gate_b[c];

  // Thread -> 4 rows x 4 cols patch. Lane n handles cols 4n..4n+3 of its wave
  // row-group => b128 LDS reads and b128 global stores.
  const int c0 = (tid & 31) << 2;          // 0,4,...,124
  const int rg = tid >> 5;                 // 0..7 (4 rows each)

  float* __restrict__ op = out + (size_t)plane * (size_t)(H_ * W_);

#pragma unroll
  for (int rr = 0; rr < 4; ++rr) {
    const int r = rg * 4 + rr;             // output row within strip, 0..31

    // center values (tile row r+1) -> gate
    const float4 xc = *(const float4*)(&tile[(r + 1) * W_ + c0]);
    const float xcv[4] = {xc.x, xc.y, xc.z, xc.w};
    float th[4];
#pragma unroll
    for (int i = 0; i < 4; ++i)
      th[i] = 1.0f / (1.0f + __expf(-(xcv[i] * gw + gb)));

    float acc[4] = {-3.402823466e38f, -3.402823466e38f,
                    -3.402823466e38f, -3.402823466e38f};
#pragma unroll
    for (int t = 0; t < 3; ++t) {          // 3 stencil rows
      const float* rp = &tile[(r + t) * W_];
      const float4 m  = *(const float4*)(rp + c0);
      const float left  = (c0 == 0)       ? 0.0f : rp[c0 - 1];
      const float right = (c0 == W_ - 4)  ? 0.0f : rp[c0 + 4];
      const float vals[6] = {left, m.x, m.y, m.z, m.w, right};
#pragma unroll
      for (int i = 0; i < 4; ++i) {
#pragma unroll
        for (int kj = 0; kj < 3; ++kj) {
          acc[i] = fmaxf(acc[i], vals[i + kj] + th[i] * kv[t * 3 + kj]);
        }
      }
    }
    *(float4*)(op + (size_t)(h0 + r) * W_ + c0) =
        make_float4(acc[0], acc[1], acc[2], acc[3]);
  }
}

extern "C" void kernel_launch(void* const* d_in, const int* in_sizes, int n_in,
                              void* d_out, int out_size, void* d_ws, size_t ws_size,
                              hipStream_t stream) {
  (void)in_sizes; (void)n_in; (void)out_size; (void)d_ws; (void)ws_size;
  const float* x      = (const float*)d_in[0];   // [B,C,H,W]
  const float* kern   = (const float*)d_in[1];   // [C,9]
  const float* gate_w = (const float*)d_in[2];   // [C]
  const float* gate_b = (const float*)d_in[3];   // [C]
  float*       out    = (float*)d_out;           // [B,C,H,W]

  dim3 grid(H_ / TH, B_ * C_);   // 4 strips x 1536 planes
  dim3 block(256);
  dynmorph_kernel<<<grid, block, 0, stream>>>(x, kern, gate_w, gate_b, out);
}